// GroupMamba_71829033058805
// MI455X (gfx1250) — compile-verified
//
#include <hip/hip_runtime.h>
#include <hip/hip_bf16.h>
#include <math.h>

typedef float v2f __attribute__((ext_vector_type(2)));
typedef float v8f __attribute__((ext_vector_type(8)));

#define BSZ 16
#define HH  32
#define WW  32
#define CCH 512
#define NTOK 1024
#define DM  128
#define LL  1024

__device__ __forceinline__ float sigmoidf_(float x) { return 1.f / (1.f + expf(-x)); }
__device__ __forceinline__ float siluf_(float x)    { return x / (1.f + expf(-x)); }
__device__ __forceinline__ float softplusf_(float x){ return (x > 20.f) ? x : log1pf(expf(x)); }

// scan index -> spatial index (H=W=32); identical mapping for input gather and output scatter
__device__ __forceinline__ int pos_of(int dir, int l) {
  switch (dir) {
    case 0:  return l;
    case 1:  return ((l & 31) << 5) | (l >> 5);
    case 2:  return 1023 - l;
    default: { int lp = 1023 - l; return ((lp & 31) << 5) | (lp >> 5); }
  }
}

// ---------------- fp32 WMMA GEMM: C[M,N] = A[M,K](lda) * W[N,K]^T (+bias) ----------------
// Each wave computes a 16x64 strip (4 adjacent 16x16 N-tiles), reusing the A fragment 4x.
// K must be a multiple of 8 (true for all uses: 128, 512). Software-pipelined double buffer.
__global__ __launch_bounds__(256) void gemm_f32_wmma(
    const float* __restrict__ A, int lda,
    const float* __restrict__ W,
    const float* __restrict__ bias,
    float* __restrict__ C, int ldc,
    int M, int N, int K)
{
  const int wave = threadIdx.x >> 5;
  const int lane = threadIdx.x & 31;
  const int tilesN = N >> 6;                       // 64-wide wave strips
  const int totalTiles = (M >> 4) * tilesN;
  const int tile = blockIdx.x * 8 + wave;          // uniform per wave
  if (tile >= totalTiles) return;                  // wave-uniform exit; EXEC stays all-1
  const int tm = tile / tilesN;
  const int tn = tile % tilesN;
  const int mrow  = (tm << 4) + (lane & 15);       // A row (M dim)
  const int ncol0 = (tn << 6) + (lane & 15);       // first N-tile column for this lane
  const int kq    = (lane >> 4) << 1;              // K pair select {0,1} or {2,3}
  const float* __restrict__ Ap = A + (size_t)mrow * lda + kq;
  const float* __restrict__ Wp = W + (size_t)ncol0 * K   + kq;
  const size_t ws = (size_t)16 * K;                // W stride between adjacent N-tiles

#define LOAD_SET(aa, q0, q1, q2, q3, kk)                                        \
  do {                                                                          \
    aa[0] = Ap[kk];            aa[1] = Ap[(kk) + 1];                            \
    q0[0] = Wp[kk];            q0[1] = Wp[(kk) + 1];                            \
    q1[0] = Wp[ws + (kk)];     q1[1] = Wp[ws + (kk) + 1];                       \
    q2[0] = Wp[2 * ws + (kk)]; q2[1] = Wp[2 * ws + (kk) + 1];                   \
    q3[0] = Wp[3 * ws + (kk)]; q3[1] = Wp[3 * ws + (kk) + 1];                   \
  } while (0)

  v8f acc0 = {0.f,0.f,0.f,0.f,0.f,0.f,0.f,0.f};
  v8f acc1 = acc0, acc2 = acc0, acc3 = acc0;
  v2f a0, p00, p01, p02, p03;                      // buffer 0
  v2f a1, p10, p11, p12, p13;                      // buffer 1
  LOAD_SET(a0, p00, p01, p02, p03, 0);
  for (int k0 = 0; k0 < K; k0 += 8) {
    LOAD_SET(a1, p10, p11, p12, p13, k0 + 4);      // in flight during first WMMA group
    acc0 = __builtin_amdgcn_wmma_f32_16x16x4_f32(false, a0, false, p00, (short)0, acc0, false, false);
    acc1 = __builtin_amdgcn_wmma_f32_16x16x4_f32(false, a0, false, p01, (short)0, acc1, false, false);
    acc2 = __builtin_amdgcn_wmma_f32_16x16x4_f32(false, a0, false, p02, (short)0, acc2, false, false);
    acc3 = __builtin_amdgcn_wmma_f32_16x16x4_f32(false, a0, false, p03, (short)0, acc3, false, false);
    if (k0 + 8 < K) {                              // uniform; no OOB prefetch on last iter
      LOAD_SET(a0, p00, p01, p02, p03, k0 + 8);    // in flight during second WMMA group
    }
    acc0 = __builtin_amdgcn_wmma_f32_16x16x4_f32(false, a1, false, p10, (short)0, acc0, false, false);
    acc1 = __builtin_amdgcn_wmma_f32_16x16x4_f32(false, a1, false, p11, (short)0, acc1, false, false);
    acc2 = __builtin_amdgcn_wmma_f32_16x16x4_f32(false, a1, false, p12, (short)0, acc2, false, false);
    acc3 = __builtin_amdgcn_wmma_f32_16x16x4_f32(false, a1, false, p13, (short)0, acc3, false, false);
  }
#undef LOAD_SET

  const int rsel = (lane >> 4) << 3;               // D rows: i + 8*(lane>>4)
  const int cbase = (tn << 6) + (lane & 15);
  float bv0 = 0.f, bv1 = 0.f, bv2 = 0.f, bv3 = 0.f;
  if (bias) {
    bv0 = bias[cbase];      bv1 = bias[cbase + 16];
    bv2 = bias[cbase + 32]; bv3 = bias[cbase + 48];
  }
#pragma unroll
  for (int i = 0; i < 8; ++i) {
    const size_t r = (size_t)((tm << 4) + i + rsel) * ldc;
    C[r + cbase]      = acc0[i] + bv0;
    C[r + cbase + 16] = acc1[i] + bv1;
    C[r + cbase + 32] = acc2[i] + bv2;
    C[r + cbase + 48] = acc3[i] + bv3;
  }
}

// ---------------- LayerNorm over 512 channels, one row per block ----------------
__global__ __launch_bounds__(256) void ln_row512(
    const float* __restrict__ x, const float* __restrict__ w,
    const float* __restrict__ b, float* __restrict__ o)
{
  __shared__ float red[256];
  const int row = blockIdx.x, t = threadIdx.x;
  const float* xr = x + (size_t)row * CCH;
  float v0 = xr[t], v1 = xr[t + 256];
  red[t] = v0 + v1; __syncthreads();
  for (int s = 128; s > 0; s >>= 1) { if (t < s) red[t] += red[t + s]; __syncthreads(); }
  float mean = red[0] * (1.f / 512.f);
  __syncthreads();
  float d0 = v0 - mean, d1 = v1 - mean;
  red[t] = d0 * d0 + d1 * d1; __syncthreads();
  for (int s = 128; s > 0; s >>= 1) { if (t < s) red[t] += red[t + s]; __syncthreads(); }
  float rstd = rsqrtf(red[0] * (1.f / 512.f) + 1e-5f);
  float* orow = o + (size_t)row * CCH;
  orow[t]       = d0 * rstd * w[t]       + b[t];
  orow[t + 256] = d1 * rstd * w[t + 256] + b[t + 256];
}

// ---------------- token mean: ZAVG[b,c] ----------------
__global__ __launch_bounds__(512) void zavg_kernel(const float* __restrict__ xn, float* __restrict__ zavg)
{
  const int b = blockIdx.x, c = threadIdx.x;
  const float* base = xn + (size_t)b * NTOK * CCH + c;
  float s = 0.f;
  for (int n = 0; n < NTOK; ++n) s += base[(size_t)n * CCH];
  zavg[b * CCH + c] = s * (1.f / (float)NTOK);
}

// ---------------- ECA + fc gate ----------------
__global__ __launch_bounds__(512) void gate_kernel(
    const float* __restrict__ zavg, const float* __restrict__ fcw,
    const float* __restrict__ fcb, const float* __restrict__ ecaw,
    float* __restrict__ gate)
{
  __shared__ float zs[CCH];
  const int b = blockIdx.x, o = threadIdx.x;
  zs[o] = zavg[b * CCH + o];
  __syncthreads();
  float acc = fcb[o];
  const float* wr = fcw + (size_t)o * CCH;
  for (int c = 0; c < CCH; ++c) acc += zs[c] * wr[c];
  float e = ecaw[1] * zs[o];
  if (o > 0)       e += ecaw[0] * zs[o - 1];
  if (o < CCH - 1) e += ecaw[2] * zs[o + 1];
  gate[b * CCH + o] = sigmoidf_(acc + e);
}

// ---------------- depthwise 3x3 conv + bias + SiLU; XZ(g,row,256) -> XC(g,row,128) ----------------
__global__ __launch_bounds__(256) void conv_silu_kernel(
    const float* __restrict__ xz, const float* __restrict__ cw,
    const float* __restrict__ cb, float* __restrict__ xc)
{
  size_t t = (size_t)blockIdx.x * 256 + threadIdx.x;  // total 4*16*32*32*128
  const int d = (int)(t & 127);
  const int w = (int)((t >> 7) & 31);
  const int h = (int)((t >> 12) & 31);
  const int b = (int)((t >> 17) & 15);
  const int g = (int)(t >> 21);
  const float* in = xz + (size_t)(g * BSZ + b) * NTOK * 256;
  const float* k  = cw + (size_t)(g * DM + d) * 9;
  float acc = cb[g * DM + d];
#pragma unroll
  for (int kh = 0; kh < 3; ++kh) {
    int hh = h + kh - 1; if (hh < 0 || hh > 31) continue;
#pragma unroll
    for (int kw = 0; kw < 3; ++kw) {
      int ww2 = w + kw - 1; if (ww2 < 0 || ww2 > 31) continue;
      acc += in[(size_t)(hh * WW + ww2) * 256 + d] * k[kh * 3 + kw];
    }
  }
  xc[((size_t)(g * BSZ + b) * NTOK + (h * WW + w)) * DM + d] = siluf_(acc);
}

// ---------------- x_proj: XD[g,b,c,l] = sum_d xs[g,b,d,l] * xpw[g,c,d]  (xs via direction gather) ----------------
__global__ __launch_bounds__(256) void xproj_kernel(
    const float* __restrict__ xc, const float* __restrict__ xpw, float* __restrict__ xd)
{
  const int t = blockIdx.x * 256 + threadIdx.x;       // (g,b,l), 65536 total
  const int l = t & 1023, b = (t >> 10) & 15, g = t >> 14;
  const int p = pos_of(g, l);
  const float* row = xc + ((size_t)(g * BSZ + b) * NTOK + p) * DM;
  const float* wp  = xpw + (size_t)g * 10 * DM;
  float acc[10];
#pragma unroll
  for (int c = 0; c < 10; ++c) acc[c] = 0.f;
  for (int dch = 0; dch < DM; ++dch) {
    float v = row[dch];
#pragma unroll
    for (int c = 0; c < 10; ++c) acc[c] += v * wp[c * DM + dch];
  }
  float* o = xd + (size_t)(g * BSZ + b) * 10 * LL;
#pragma unroll
  for (int c = 0; c < 10; ++c) o[c * LL + l] = acc[c];
}

// ---------------- selective scan: fused dt_proj+softplus+recurrence; y scattered to spatial order ----------------
__global__ __launch_bounds__(256) void scan_kernel(
    const float* __restrict__ xc, const float* __restrict__ xd,
    const float* __restrict__ dtw, const float* __restrict__ dtb,
    const float* __restrict__ Alog, const float* __restrict__ Dp,
    float* __restrict__ y)
{
  const int t = blockIdx.x * 256 + threadIdx.x;       // (g,b,d), 8192 total
  const int d = t & 127, b = (t >> 7) & 15, g = t >> 11;
  const float* xcb = xc + (size_t)(g * BSZ + b) * NTOK * DM;
  const float* xdb = xd + (size_t)(g * BSZ + b) * 10 * LL;
  float* yb = y + (size_t)(g * BSZ + b) * NTOK * DM;
  float w[8];
#pragma unroll
  for (int r = 0; r < 8; ++r) w[r] = dtw[(g * DM + d) * 8 + r];
  const float bdt = dtb[g * DM + d];
  const float A   = -expf(Alog[g * DM + d]);
  const float Dd  = Dp[g * DM + d];
  float h = 0.f;
  for (int l = 0; l < LL; ++l) {
    const int p = pos_of(g, l);
    const float u = xcb[(size_t)p * DM + d];
    float s = bdt;
#pragma unroll
    for (int r = 0; r < 8; ++r) s += xdb[r * LL + l] * w[r];
    const float dt = softplusf_(s);
    const float Bc = xdb[8 * LL + l];
    const float Cc = xdb[9 * LL + l];
    h = expf(dt * A) * h + dt * Bc * u;
    yb[(size_t)p * DM + d] = h * Cc + Dd * u;
  }
}

// ---------------- out_norm (LN over 128) * SiLU(z) ----------------
__global__ __launch_bounds__(128) void outnorm_gate_kernel(
    const float* __restrict__ y, const float* __restrict__ xz,
    const float* __restrict__ onw, const float* __restrict__ onb,
    float* __restrict__ yo)
{
  __shared__ float red[128];
  const int idx = blockIdx.x;                 // (g*16+b)*1024+n
  const int t = threadIdx.x;
  const float v = y[(size_t)idx * DM + t];
  red[t] = v; __syncthreads();
  for (int s = 64; s > 0; s >>= 1) { if (t < s) red[t] += red[t + s]; __syncthreads(); }
  float mean = red[0] * (1.f / 128.f);
  __syncthreads();
  float dv = v - mean;
  red[t] = dv * dv; __syncthreads();
  for (int s = 64; s > 0; s >>= 1) { if (t < s) red[t] += red[t + s]; __syncthreads(); }
  float rstd = rsqrtf(red[0] * (1.f / 128.f) + 1e-5f);
  const int g = idx >> 14;
  const float z = xz[(size_t)idx * 256 + 128 + t];
  yo[(size_t)idx * DM + t] = (dv * rstd * onw[g * DM + t] + onb[g * DM + t]) * siluf_(z);
}

// ---------------- combine: xm = ys*skip*xn*gate, then LN(norm_w,norm_b) ----------------
__global__ __launch_bounds__(256) void combine_ln_kernel(
    const float* __restrict__ ymix, const float* __restrict__ xn,
    const float* __restrict__ gate, const float* __restrict__ skip,
    const float* __restrict__ w, const float* __restrict__ bb,
    float* __restrict__ o)
{
  __shared__ float red[256];
  const int row = blockIdx.x, t = threadIdx.x;
  const int b = row >> 10;
  const float ss = skip[0];
  const size_t base = (size_t)row * CCH;
  float v0 = ymix[base + t]       * ss * xn[base + t]       * gate[b * CCH + t];
  float v1 = ymix[base + t + 256] * ss * xn[base + t + 256] * gate[b * CCH + t + 256];
  red[t] = v0 + v1; __syncthreads();
  for (int s = 128; s > 0; s >>= 1) { if (t < s) red[t] += red[t + s]; __syncthreads(); }
  float mean = red[0] * (1.f / 512.f);
  __syncthreads();
  float d0 = v0 - mean, d1 = v1 - mean;
  red[t] = d0 * d0 + d1 * d1; __syncthreads();
  for (int s = 128; s > 0; s >>= 1) { if (t < s) red[t] += red[t + s]; __syncthreads(); }
  float rstd = rsqrtf(red[0] * (1.f / 512.f) + 1e-5f);
  o[base + t]       = d0 * rstd * w[t]       + bb[t];
  o[base + t + 256] = d1 * rstd * w[t + 256] + bb[t + 256];
}

extern "C" void kernel_launch(void* const* d_in, const int* in_sizes, int n_in,
                              void* d_out, int out_size, void* d_ws, size_t ws_size,
                              hipStream_t stream)
{
  const float* x      = (const float*)d_in[0];
  const float* norm_w = (const float*)d_in[3];
  const float* norm_b = (const float*)d_in[4];
  const float* fc_w   = (const float*)d_in[5];
  const float* fc_b   = (const float*)d_in[6];
  const float* eca_w  = (const float*)d_in[7];
  const float* skip   = (const float*)d_in[8];
  const float* proj_w = (const float*)d_in[9];
  const float* proj_b = (const float*)d_in[10];
  const float* ipw    = (const float*)d_in[11];
  const float* cw     = (const float*)d_in[12];
  const float* cb     = (const float*)d_in[13];
  const float* xpw    = (const float*)d_in[14];
  const float* dtw    = (const float*)d_in[15];
  const float* dtb    = (const float*)d_in[16];
  const float* Alog   = (const float*)d_in[17];
  const float* Dp     = (const float*)d_in[18];
  const float* onw    = (const float*)d_in[19];
  const float* onb    = (const float*)d_in[20];
  const float* opw    = (const float*)d_in[21];
  float* out = (float*)d_out;
  float* ws  = (float*)d_ws;

  const size_t M = (size_t)BSZ * NTOK;     // 16384 rows
  float* XN   = ws;                        //  8,388,608 f  (B,N,512)
  float* XZ   = ws + 8388608;              // 16,777,216 f  (4,rows,256)
  float* XC   = ws + 25165824;             //  8,388,608 f  (4,rows,128)
  float* Y    = ws + 33554432;             //  8,388,608 f  (4,rows,128) / later (rows,512)
  float* XD   = ws + 41943040;             //    655,360 f  (4,B,10,L)
  float* ZAVG = ws + 42598400;             //      8,192 f
  float* GATE = ws + 42606592;             //      8,192 f
  float* XMN  = XZ;                        // reuse XZ after z consumed

  // 1) LayerNorm(x) -> XN
  ln_row512<<<(int)M, 256, 0, stream>>>(x, norm_w, norm_b, XN);
  // 2) token mean -> ZAVG
  zavg_kernel<<<BSZ, 512, 0, stream>>>(XN, ZAVG);
  // 3) ECA + fc gate -> GATE
  gate_kernel<<<BSZ, 512, 0, stream>>>(ZAVG, fc_w, fc_b, eca_w, GATE);
  // 4) in_proj GEMMs (WMMA): (16384x128) x (256x128)^T per group; 1024*(256/64)=4096 strips
  for (int g = 0; g < 4; ++g) {
    gemm_f32_wmma<<<512, 256, 0, stream>>>(
        XN + g * 128, CCH,
        ipw + (size_t)g * 256 * 128, nullptr,
        XZ + (size_t)g * M * 256, 256,
        (int)M, 256, 128);
  }
  // 5) depthwise conv + SiLU -> XC
  conv_silu_kernel<<<32768, 256, 0, stream>>>(XZ, cw, cb, XC);
  // 6) x_proj -> XD
  xproj_kernel<<<256, 256, 0, stream>>>(XC, xpw, XD);
  // 7) fused dt_proj + softplus + selective scan -> Y (spatial order)
  scan_kernel<<<32, 256, 0, stream>>>(XC, XD, dtw, dtb, Alog, Dp, Y);
  // 8) out_norm * SiLU(z) -> XC (reused as yo)
  outnorm_gate_kernel<<<65536, 128, 0, stream>>>(Y, XZ, onw, onb, XC);
  // 9) out_proj GEMMs (WMMA) -> Y viewed as (rows,512), group g at col g*128; 1024*2=2048 strips
  for (int g = 0; g < 4; ++g) {
    gemm_f32_wmma<<<256, 256, 0, stream>>>(
        XC + (size_t)g * M * 128, 128,
        opw + (size_t)g * 128 * 128, nullptr,
        Y + g * 128, CCH,
        (int)M, 128, 128);
  }
  // 10) skip*xh*gate + LayerNorm -> XMN
  combine_ln_kernel<<<(int)M, 256, 0, stream>>>(Y, XN, GATE, skip, norm_w, norm_b, XMN);
  // 11) final proj (WMMA) + bias -> out; 1024*(512/64)=8192 strips
  gemm_f32_wmma<<<1024, 256, 0, stream>>>(XMN, CCH, proj_w, proj_b, out, CCH, (int)M, CCH, CCH);
}